// quant_lookup_4707284156810
// MI455X (gfx1250) — compile-verified
//
#include <hip/hip_runtime.h>
#include <math.h>

// LLT activation quantizer forward: out = ((lut[idx] + g) - g) * sc
//   g   = clip(x/sc, -1, 1)
//   idx = clip(rint(g*120) + 120, 0, 240)
//   lut = [zeros(121), cumsum(softmax(table/tau, axis=1).flatten())] / 15
// Histogram/wgt only affect gradients -> skipped. Lazy scale init handled by a
// dynamically-skipped std reduction (scale[0] != 0 for the given input).

#define QRANGE 15
#define QGRANU 8
#define QL     120          // RANGE*GRANU
#define QTABN  (2*QL + 1)   // 241

typedef float v4f __attribute__((ext_vector_type(4)));

struct LltWs {
    double sum;      // offset 0
    double sumsq;    // offset 8
    float  sc;       // offset 16
    float  pad[3];
    float  lut[QTABN];
};

// ---------------------------------------------------------------- init
__global__ void llt_zero_ws(LltWs* ws) {
    if (threadIdx.x == 0) { ws->sum = 0.0; ws->sumsq = 0.0; }
}

// ------------------------------------------------- std reduction (lazy init)
// Only does real work when scale[0]==0 (lazy-init branch); otherwise every
// block early-exits after one broadcast load -> ~free for the given input.
__global__ __launch_bounds__(256) void llt_reduce_std(
    const float* __restrict__ x, const float* __restrict__ scale,
    long long n4, long long n, LltWs* __restrict__ ws)
{
    if (scale[0] != 0.0f) return;

    const v4f* __restrict__ x4 = (const v4f*)x;
    double s = 0.0, ss = 0.0;
    long long stride = (long long)gridDim.x * blockDim.x;
    for (long long i = (long long)blockIdx.x * blockDim.x + threadIdx.x; i < n4; i += stride) {
        v4f v = x4[i];
        s  += (double)v.x + (double)v.y + (double)v.z + (double)v.w;
        ss += (double)v.x * v.x + (double)v.y * v.y
            + (double)v.z * v.z + (double)v.w * v.w;
    }
    for (long long i = 4 * n4 + (long long)blockIdx.x * blockDim.x + threadIdx.x; i < n; i += stride) {
        double v = (double)x[i];
        s += v; ss += v * v;
    }

    __shared__ double r0[256];
    __shared__ double r1[256];
    int t = threadIdx.x;
    r0[t] = s; r1[t] = ss;
    __syncthreads();
    for (int off = 128; off > 0; off >>= 1) {
        if (t < off) { r0[t] += r0[t + off]; r1[t] += r1[t + off]; }
        __syncthreads();
    }
    if (t == 0) {
        atomicAdd(&ws->sum,   r0[0]);
        atomicAdd(&ws->sumsq, r1[0]);
    }
}

// ------------------------------------------------------------ LUT build
// softmax per row (15x8), serial cumsum over 120 (matches jnp.cumsum order),
// /15, with 121 leading zeros. Also resolves sc = exp(scale or log(3*std)).
__global__ __launch_bounds__(256) void llt_build_lut(
    const float* __restrict__ table, const float* __restrict__ scale,
    const float* __restrict__ tau, long long n, LltWs* __restrict__ ws)
{
    __shared__ float prob[QL];
    int t = threadIdx.x;

    if (t == 0) {
        float sp = scale[0];
        if (sp == 0.0f) {   // lazy init: log(3 * std(x, ddof=1))
            double sum = ws->sum, ssq = ws->sumsq;
            double mean = sum / (double)n;
            double var  = (ssq - sum * mean) / (double)(n - 1);
            if (var < 0.0) var = 0.0;
            sp = logf((float)sqrt(var) * 3.0f);
        }
        ws->sc = expf(sp);
    }

    if (t < QRANGE) {
        float tv = tau[0];
        float z[QGRANU];
        float m = -INFINITY;
        #pragma unroll
        for (int j = 0; j < QGRANU; ++j) {
            z[j] = table[t * QGRANU + j] / tv;
            m = fmaxf(m, z[j]);
        }
        float sum = 0.0f;
        #pragma unroll
        for (int j = 0; j < QGRANU; ++j) { z[j] = expf(z[j] - m); sum += z[j]; }
        #pragma unroll
        for (int j = 0; j < QGRANU; ++j) prob[t * QGRANU + j] = z[j] / sum;
    }
    __syncthreads();

    if (t == 0) {           // serial cumsum, canonical left-to-right order
        float acc = 0.0f;
        for (int i = 0; i < QL; ++i) { acc += prob[i]; prob[i] = acc; }
    }
    __syncthreads();

    for (int i = t; i < QTABN; i += blockDim.x)
        ws->lut[i] = (i <= QL) ? 0.0f : prob[i - QL - 1] / 15.0f;
}

// ------------------------------------------------------------- main pass
__device__ __forceinline__ float llt_quant1(float xv, float sc, const float* lut) {
    float g = fminf(fmaxf(xv / sc, -1.0f), 1.0f);   // clip(x/sc,-1,1)
    int idx = (int)rintf(g * 120.0f) + 120;         // round-half-even, like jnp
    idx = idx < 0 ? 0 : (idx > 2 * QL ? 2 * QL : idx);
    float tq = lut[idx];                            // ds_load_b32 gather
    return ((tq + g) - g) * sc;                     // exact ref eval order
}

__global__ __launch_bounds__(256) void llt_lookup(
    const float* __restrict__ x, float* __restrict__ out,
    const LltWs* __restrict__ ws, long long n4, long long n)
{
    __shared__ float s_lut[QTABN];

    if (threadIdx.x < QTABN) {
#if defined(__gfx1250__)
        // gfx1250 async global->LDS broadcast of the LUT (tracked by ASYNCcnt)
        unsigned lds_off = (unsigned)(unsigned long long)(void*)&s_lut[threadIdx.x];
        unsigned long long ga = (unsigned long long)(const void*)&ws->lut[threadIdx.x];
        asm volatile("global_load_async_to_lds_b32 %0, %1, off"
                     :: "v"(lds_off), "v"(ga) : "memory");
#else
        s_lut[threadIdx.x] = ws->lut[threadIdx.x];
#endif
    }
    float sc = ws->sc;   // uniform scalar load
#if defined(__gfx1250__)
    asm volatile("s_wait_asynccnt 0" ::: "memory");
#endif
    __syncthreads();

    const v4f* __restrict__ x4 = (const v4f*)x;
    v4f* __restrict__ o4 = (v4f*)out;
    long long stride = (long long)gridDim.x * blockDim.x;

    // Streaming b128 nontemporal loads/stores: 411 MB total > 192 MB L2,
    // so bypassing the temporal path is the right policy.
    for (long long i = (long long)blockIdx.x * blockDim.x + threadIdx.x; i < n4; i += stride) {
        v4f v = __builtin_nontemporal_load(x4 + i);
        v4f r;
        r.x = llt_quant1(v.x, sc, s_lut);
        r.y = llt_quant1(v.y, sc, s_lut);
        r.z = llt_quant1(v.z, sc, s_lut);
        r.w = llt_quant1(v.w, sc, s_lut);
        __builtin_nontemporal_store(r, o4 + i);
    }
    for (long long i = 4 * n4 + (long long)blockIdx.x * blockDim.x + threadIdx.x; i < n; i += stride)
        out[i] = llt_quant1(x[i], sc, s_lut);
}

// ---------------------------------------------------------------- launch
extern "C" void kernel_launch(void* const* d_in, const int* in_sizes, int n_in,
                              void* d_out, int out_size, void* d_ws, size_t ws_size,
                              hipStream_t stream) {
    const float* x     = (const float*)d_in[0];
    const float* table = (const float*)d_in[1];
    const float* scale = (const float*)d_in[2];
    const float* tau   = (const float*)d_in[3];
    float* out = (float*)d_out;
    LltWs* ws  = (LltWs*)d_ws;

    long long n  = (long long)in_sizes[0];
    long long n4 = n / 4;

    llt_zero_ws<<<1, 64, 0, stream>>>(ws);

    long long rb = (n4 + 255) / 256;
    int rblocks = (int)(rb < 1 ? 1 : (rb > 1024 ? 1024 : rb));
    llt_reduce_std<<<rblocks, 256, 0, stream>>>(x, scale, n4, n, ws);

    llt_build_lut<<<1, 256, 0, stream>>>(table, scale, tau, n, ws);

    long long mb = (n4 + 1023) / 1024;   // 4 x float4 per thread per sweep
    int mblocks = (int)(mb < 1 ? 1 : mb);
    llt_lookup<<<mblocks, 256, 0, stream>>>(x, out, ws, n4, n);
}